// Metalayer_sub_62869731279045
// MI455X (gfx1250) — compile-verified
//
#include <hip/hip_runtime.h>
#include <math.h>

// ---------------------------------------------------------------------------
// Metalayer pipeline for MI455X (gfx1250, wave32, WMMA).
// Heavy path: Newton-Schulz inverse + scaled cos/sin Taylor + 8 complex
// squarings, all driven through one fused FP32 WMMA GEMM:
//     D = alpha * A@B + beta * Cin + diagAdd * I     (n x n, n % 128 == 0)
// GEMM: double-buffered LDS pipeline, K-pair-interleaved LDS layout so each
// WMMA fragment is a single aligned ds_load_b64 (no v_mov packing), plus
// global_prefetch two stages ahead.
// ---------------------------------------------------------------------------

#define HD 64      // MLP hidden width
#define RESN 32    // field resolution per waveguide

typedef float v2f __attribute__((ext_vector_type(2)));
typedef float v8f __attribute__((ext_vector_type(8)));

#define LDSTRIDE 258   // floats per kp-row (2*128 + 2 pad), even -> 8B aligned

// ============================ WMMA FP32 GEMM ===============================
// Block tile 128x128, BK=16. 256 threads = 8 waves arranged 4(M) x 2(N).
// Each wave computes 32x64 = 2x4 tiles of 16x16 via V_WMMA_F32_16X16X4_F32.
// LDS layout: Asp[buf][kp][2*m + j]  = A[row0+m][kb + 2*kp + j]
//             Bsp[buf][kp][2*c + j]  = B[kb + 2*kp + j][col0 + c]
__global__ __launch_bounds__(256)
void gemm_f32_wmma(const float* __restrict__ A, const float* __restrict__ B,
                   const float* __restrict__ Cin, float* __restrict__ D,
                   int n, float alpha, float beta, float diagAdd)
{
    __shared__ float Asp[2][8][LDSTRIDE];
    __shared__ float Bsp[2][8][LDSTRIDE];

    const int tid   = threadIdx.x;
    const int lane  = tid & 31;
    const int wid   = tid >> 5;
    const int waveM = wid >> 1;         // 0..3  -> 32 rows each
    const int waveN = wid & 1;          // 0..1  -> 64 cols each
    const int half  = lane >> 4;        // lane group (0: lanes 0-15, 1: 16-31)
    const int l15   = lane & 15;

    const int row0 = blockIdx.y * 128;
    const int col0 = blockIdx.x * 128;

    // staging assignments
    const int am = tid >> 1;            // A: row within tile, 0..127
    const int akg = (tid & 1) * 8;      // A: k-offset 0 or 8 (8 consecutive k)
    const int bkp = tid >> 5;           // B: k-pair 0..7
    const int bcg = (tid & 31) * 4;     // B: col group 0..124

    const float* aptr = A + (size_t)(row0 + am) * n + akg;
    const float* bptr0 = B + (size_t)(2 * bkp) * n + col0 + bcg;
    const float* bptr1 = bptr0 + n;

    v8f acc[2][4] = {};                 // zero-init accumulators

    // ---- prologue: stage kb = 0 into buffer 0 ----
    float4 ra0 = *(const float4*)(aptr);
    float4 ra1 = *(const float4*)(aptr + 4);
    float4 rb0 = *(const float4*)(bptr0);
    float4 rb1 = *(const float4*)(bptr1);
    {
        float2 t;
        t.x = ra0.x; t.y = ra0.y; *(float2*)&Asp[0][(akg >> 1) + 0][2 * am] = t;
        t.x = ra0.z; t.y = ra0.w; *(float2*)&Asp[0][(akg >> 1) + 1][2 * am] = t;
        t.x = ra1.x; t.y = ra1.y; *(float2*)&Asp[0][(akg >> 1) + 2][2 * am] = t;
        t.x = ra1.z; t.y = ra1.w; *(float2*)&Asp[0][(akg >> 1) + 3][2 * am] = t;
        t.x = rb0.x; t.y = rb1.x; *(float2*)&Bsp[0][bkp][2 * (bcg + 0)] = t;
        t.x = rb0.y; t.y = rb1.y; *(float2*)&Bsp[0][bkp][2 * (bcg + 1)] = t;
        t.x = rb0.z; t.y = rb1.z; *(float2*)&Bsp[0][bkp][2 * (bcg + 2)] = t;
        t.x = rb0.w; t.y = rb1.w; *(float2*)&Bsp[0][bkp][2 * (bcg + 3)] = t;
    }
    __syncthreads();

    int buf = 0;
    for (int kb = 0; kb < n; kb += 16) {
        const bool hasNext = (kb + 16) < n;
        // issue next stage's global loads early (latency hidden behind WMMA)
        if (hasNext) {
            ra0 = *(const float4*)(aptr + kb + 16);
            ra1 = *(const float4*)(aptr + kb + 16 + 4);
            rb0 = *(const float4*)(bptr0 + (size_t)(kb + 16) * n);
            rb1 = *(const float4*)(bptr1 + (size_t)(kb + 16) * n);
        }
        // prefetch two stages ahead into the cache hierarchy
        if ((kb + 32) < n) {
            __builtin_prefetch(aptr + kb + 32, 0, 3);
            __builtin_prefetch(bptr0 + (size_t)(kb + 32) * n, 0, 3);
        }

        // ---- compute current stage: 4 k-steps of K=4 ----
        #pragma unroll
        for (int kk4 = 0; kk4 < 4; ++kk4) {
            const int kp = kk4 * 2 + half;   // K = 2*kp + {0,1} = 4*kk4 + 2*half + j
            v2f afrag[2];
            #pragma unroll
            for (int tm = 0; tm < 2; ++tm) {
                const int m = waveM * 32 + tm * 16 + l15;
                afrag[tm] = *(const v2f*)&Asp[buf][kp][2 * m];   // ds_load_b64
            }
            v2f bfrag[4];
            #pragma unroll
            for (int tn = 0; tn < 4; ++tn) {
                const int c = waveN * 64 + tn * 16 + l15;
                bfrag[tn] = *(const v2f*)&Bsp[buf][kp][2 * c];   // ds_load_b64
            }
            #pragma unroll
            for (int tm = 0; tm < 2; ++tm)
                #pragma unroll
                for (int tn = 0; tn < 4; ++tn)
                    acc[tm][tn] = __builtin_amdgcn_wmma_f32_16x16x4_f32(
                        false, afrag[tm], false, bfrag[tn],
                        (short)0, acc[tm][tn], false, false);
        }

        // ---- store next stage into the other buffer ----
        if (hasNext) {
            const int nb = buf ^ 1;
            float2 t;
            t.x = ra0.x; t.y = ra0.y; *(float2*)&Asp[nb][(akg >> 1) + 0][2 * am] = t;
            t.x = ra0.z; t.y = ra0.w; *(float2*)&Asp[nb][(akg >> 1) + 1][2 * am] = t;
            t.x = ra1.x; t.y = ra1.y; *(float2*)&Asp[nb][(akg >> 1) + 2][2 * am] = t;
            t.x = ra1.z; t.y = ra1.w; *(float2*)&Asp[nb][(akg >> 1) + 3][2 * am] = t;
            t.x = rb0.x; t.y = rb1.x; *(float2*)&Bsp[nb][bkp][2 * (bcg + 0)] = t;
            t.x = rb0.y; t.y = rb1.y; *(float2*)&Bsp[nb][bkp][2 * (bcg + 1)] = t;
            t.x = rb0.z; t.y = rb1.z; *(float2*)&Bsp[nb][bkp][2 * (bcg + 2)] = t;
            t.x = rb0.w; t.y = rb1.w; *(float2*)&Bsp[nb][bkp][2 * (bcg + 3)] = t;
        }
        __syncthreads();
        buf ^= 1;
    }

    // ---- epilogue: D = alpha*acc + beta*Cin + diagAdd*I ----
    // C/D layout: VGPR r holds M=r (lanes 0-15) and M=r+8 (lanes 16-31), N=l15.
    #pragma unroll
    for (int tm = 0; tm < 2; ++tm) {
        #pragma unroll
        for (int tn = 0; tn < 4; ++tn) {
            const int col = col0 + waveN * 64 + tn * 16 + l15;
            #pragma unroll
            for (int r = 0; r < 8; ++r) {
                const int row = row0 + waveM * 32 + tm * 16 + r + half * 8;
                const size_t idx = (size_t)row * n + col;
                float v = alpha * acc[tm][tn][r];
                if (Cin) v += beta * Cin[idx];
                if (row == col) v += diagAdd;
                D[idx] = v;
            }
        }
    }
}

// ============================ small kernels ================================

// neffs = 2 + 0.5*tanh(MLP3_{1->64->64->1}(hs))
__global__ __launch_bounds__(256)
void neff_kernel(const float* __restrict__ hs,
                 const float* __restrict__ W1, const float* __restrict__ b1,
                 const float* __restrict__ W2, const float* __restrict__ b2,
                 const float* __restrict__ W3, const float* __restrict__ b3,
                 float* __restrict__ neffs, int N)
{
    __shared__ float sW1[HD], sb1[HD], sW2[HD * HD], sb2[HD], sW3[HD], sb3;
    for (int i = threadIdx.x; i < HD; i += 256) {
        sW1[i] = W1[i]; sb1[i] = b1[i]; sb2[i] = b2[i]; sW3[i] = W3[i];
    }
    for (int i = threadIdx.x; i < HD * HD; i += 256) sW2[i] = W2[i];
    if (threadIdx.x == 0) sb3 = b3[0];
    __syncthreads();

    const int m = blockIdx.x * 256 + threadIdx.x;
    if (m >= N) return;
    const float x = hs[m];
    float h1[HD], h2[HD];
    #pragma unroll 8
    for (int j = 0; j < HD; ++j) h1[j] = fmaxf(x * sW1[j] + sb1[j], 0.f);
    for (int j = 0; j < HD; ++j) {
        float s = sb2[j];
        #pragma unroll 8
        for (int i = 0; i < HD; ++i) s += h1[i] * sW2[i * HD + j];
        h2[j] = fmaxf(s, 0.f);
    }
    float o = sb3;
    #pragma unroll 8
    for (int i = 0; i < HD; ++i) o += h2[i] * sW3[i];
    neffs[m] = 2.0f + 0.5f * tanhf(o);
}

// Eys = MLP3_{1->64->64->32}(hs);  U0 = C_EPS*DX * <Eys_row, E0c_row>
__global__ __launch_bounds__(256)
void eys_kernel(const float* __restrict__ hs,
                const float* __restrict__ W1, const float* __restrict__ b1,
                const float* __restrict__ W2, const float* __restrict__ b2,
                const float* __restrict__ W3, const float* __restrict__ b3,
                const float* __restrict__ E0, int e0_off,
                float* __restrict__ Eys, float* __restrict__ U0,
                int N, float ceps_dx)
{
    __shared__ float sW1[HD], sb1[HD], sW2[HD * HD], sb2[HD],
                     sW3[HD * RESN], sb3[RESN];
    for (int i = threadIdx.x; i < HD; i += 256) {
        sW1[i] = W1[i]; sb1[i] = b1[i]; sb2[i] = b2[i];
    }
    for (int i = threadIdx.x; i < HD * HD; i += 256) sW2[i] = W2[i];
    for (int i = threadIdx.x; i < HD * RESN; i += 256) sW3[i] = W3[i];
    for (int i = threadIdx.x; i < RESN; i += 256) sb3[i] = b3[i];
    __syncthreads();

    const int m = blockIdx.x * 256 + threadIdx.x;
    if (m >= N) return;
    const float x = hs[m];
    float h1[HD], h2[HD];
    #pragma unroll 8
    for (int j = 0; j < HD; ++j) h1[j] = fmaxf(x * sW1[j] + sb1[j], 0.f);
    for (int j = 0; j < HD; ++j) {
        float s = sb2[j];
        #pragma unroll 8
        for (int i = 0; i < HD; ++i) s += h1[i] * sW2[i * HD + j];
        h2[j] = fmaxf(s, 0.f);
    }
    float u0 = 0.f;
    for (int r = 0; r < RESN; ++r) {
        float o = sb3[r];
        #pragma unroll 8
        for (int i = 0; i < HD; ++i) o += h2[i] * sW3[i * RESN + r];
        Eys[(size_t)m * RESN + r] = o;
        u0 += o * E0[e0_off + (size_t)m * RESN + r];
    }
    U0[m] = ceps_dx * u0;
}

// C = I (scatter target), Km = 0
__global__ __launch_bounds__(256)
void init_ck_kernel(float* __restrict__ C, float* __restrict__ Km, int n)
{
    const long long i = (long long)blockIdx.x * 256 + threadIdx.x;
    if (i >= (long long)n * n) return;
    const int row = (int)(i / n);
    const int col = (int)(i - (long long)row * n);
    C[i]  = (row == col) ? 1.f : 0.f;
    Km[i] = 0.f;
}

// per-edge MLPs (3->64->64->1) for c_e and k_e, scatter into C and Km
__global__ __launch_bounds__(256)
void edge_kernel(const float* __restrict__ hs, const float* __restrict__ dis,
                 const int* __restrict__ src, const int* __restrict__ dst,
                 const float* __restrict__ cW1, const float* __restrict__ cb1,
                 const float* __restrict__ cW2, const float* __restrict__ cb2,
                 const float* __restrict__ cW3, const float* __restrict__ cb3,
                 const float* __restrict__ kW1, const float* __restrict__ kb1,
                 const float* __restrict__ kW2, const float* __restrict__ kb2,
                 const float* __restrict__ kW3, const float* __restrict__ kb3,
                 float* __restrict__ C, float* __restrict__ Km,
                 int N, int E, float kwave)
{
    __shared__ float scW1[3 * HD], scb1[HD], scW2[HD * HD], scb2[HD], scW3[HD], scb3;
    __shared__ float skW1[3 * HD], skb1[HD], skW2[HD * HD], skb2[HD], skW3[HD], skb3;
    for (int i = threadIdx.x; i < 3 * HD; i += 256) { scW1[i] = cW1[i]; skW1[i] = kW1[i]; }
    for (int i = threadIdx.x; i < HD; i += 256) {
        scb1[i] = cb1[i]; scb2[i] = cb2[i]; scW3[i] = cW3[i];
        skb1[i] = kb1[i]; skb2[i] = kb2[i]; skW3[i] = kW3[i];
    }
    for (int i = threadIdx.x; i < HD * HD; i += 256) { scW2[i] = cW2[i]; skW2[i] = kW2[i]; }
    if (threadIdx.x == 0) { scb3 = cb3[0]; skb3 = kb3[0]; }
    __syncthreads();

    const int e = blockIdx.x * 256 + threadIdx.x;
    if (e >= E) return;
    const int s = src[e], d = dst[e];
    const float x0 = hs[s], x1 = hs[d], x2 = dis[e];

    float h1[HD], h2[HD];
    // ---- c net ----
    #pragma unroll 8
    for (int j = 0; j < HD; ++j)
        h1[j] = fmaxf(x0 * scW1[j] + x1 * scW1[HD + j] + x2 * scW1[2 * HD + j] + scb1[j], 0.f);
    for (int j = 0; j < HD; ++j) {
        float sacc = scb2[j];
        #pragma unroll 8
        for (int i = 0; i < HD; ++i) sacc += h1[i] * scW2[i * HD + j];
        h2[j] = fmaxf(sacc, 0.f);
    }
    float oc = scb3;
    #pragma unroll 8
    for (int i = 0; i < HD; ++i) oc += h2[i] * scW3[i];
    const float c_e = 0.1f * tanhf(oc);
    // ---- k net ----
    #pragma unroll 8
    for (int j = 0; j < HD; ++j)
        h1[j] = fmaxf(x0 * skW1[j] + x1 * skW1[HD + j] + x2 * skW1[2 * HD + j] + skb1[j], 0.f);
    for (int j = 0; j < HD; ++j) {
        float sacc = skb2[j];
        #pragma unroll 8
        for (int i = 0; i < HD; ++i) sacc += h1[i] * skW2[i * HD + j];
        h2[j] = fmaxf(sacc, 0.f);
    }
    float ok = skb3;
    #pragma unroll 8
    for (int i = 0; i < HD; ++i) ok += h2[i] * skW3[i];
    const float k_e = 0.1f * kwave * tanhf(ok);

    // (src,dst) pairs are unique (distinct nonzero offsets) -> race-free
    const size_t idx = (size_t)s * N + d;
    C[idx]  += c_e;
    Km[idx] += k_e;
}

// D = a*X + b*Y + diag*I  (X/Y nullable)
__global__ __launch_bounds__(256)
void elem_kernel(float* __restrict__ D, const float* __restrict__ X,
                 const float* __restrict__ Y, float a, float b, float diag, int n)
{
    const long long i = (long long)blockIdx.x * 256 + threadIdx.x;
    if (i >= (long long)n * n) return;
    const int row = (int)(i / n);
    const int col = (int)(i - (long long)row * n);
    float v = 0.f;
    if (X) v += a * X[i];
    if (Y) v += b * Y[i];
    if (row == col) v += diag;
    D[i] = v;
}

// M1 = diag(neffs*kwave) * C + Km
__global__ __launch_bounds__(256)
void m1_kernel(float* __restrict__ D, const float* __restrict__ C,
               const float* __restrict__ Km, const float* __restrict__ neffs,
               float kwave, int n)
{
    const long long i = (long long)blockIdx.x * 256 + threadIdx.x;
    if (i >= (long long)n * n) return;
    const int row = (int)(i / n);
    D[i] = neffs[row] * kwave * C[i] + Km[i];
}

// UzRe = X@u, UzIm = Y@u   (one wave per row)
__global__ __launch_bounds__(256)
void matvec2_kernel(const float* __restrict__ X, const float* __restrict__ Y,
                    const float* __restrict__ u, float* __restrict__ outRe,
                    float* __restrict__ outIm, int n)
{
    const int lane = threadIdx.x & 31;
    const int wid  = threadIdx.x >> 5;
    const int row  = blockIdx.x * 8 + wid;
    if (row >= n) return;
    const float* xr = X + (size_t)row * n;
    const float* yr = Y + (size_t)row * n;
    float sx = 0.f, sy = 0.f;
    for (int k = lane; k < n; k += 32) {
        const float uv = u[k];
        sx += xr[k] * uv;
        sy += yr[k] * uv;
    }
    #pragma unroll
    for (int off = 16; off > 0; off >>= 1) {
        sx += __shfl_xor(sx, off, 32);
        sy += __shfl_xor(sy, off, 32);
    }
    if (lane == 0) { outRe[row] = sx; outIm[row] = sy; }
}

// En[m*RES+r] = Uz[m] * Eys[m][r]  -> interleaved complex64 floats
__global__ __launch_bounds__(256)
void en_kernel(const float* __restrict__ UzRe, const float* __restrict__ UzIm,
               const float* __restrict__ Eys, float* __restrict__ out, int total)
{
    const int t = blockIdx.x * 256 + threadIdx.x;
    if (t >= total) return;
    const int m = t >> 5;   // RES = 32
    const float ey = Eys[t];
    out[2 * t]     = UzRe[m] * ey;
    out[2 * t + 1] = UzIm[m] * ey;
}

// ============================ host orchestration ===========================

static inline void gemm(const float* A, const float* B, const float* Cin, float* D,
                        int n, float alpha, float beta, float diag, hipStream_t st)
{
    dim3 grid(n / 128, n / 128);
    gemm_f32_wmma<<<grid, dim3(256), 0, st>>>(A, B, Cin, D, n, alpha, beta, diag);
}

extern "C" void kernel_launch(void* const* d_in, const int* in_sizes, int n_in,
                              void* d_out, int out_size, void* d_ws, size_t ws_size,
                              hipStream_t stream)
{
    const float* E0  = (const float*)d_in[0];
    const float* hs  = (const float*)d_in[1];
    const float* dis = (const float*)d_in[2];
    const float* nW1 = (const float*)d_in[3];  const float* nb1 = (const float*)d_in[4];
    const float* nW2 = (const float*)d_in[5];  const float* nb2 = (const float*)d_in[6];
    const float* nW3 = (const float*)d_in[7];  const float* nb3 = (const float*)d_in[8];
    const float* cW1 = (const float*)d_in[9];  const float* cb1 = (const float*)d_in[10];
    const float* cW2 = (const float*)d_in[11]; const float* cb2 = (const float*)d_in[12];
    const float* cW3 = (const float*)d_in[13]; const float* cb3 = (const float*)d_in[14];
    const float* kW1 = (const float*)d_in[15]; const float* kb1 = (const float*)d_in[16];
    const float* kW2 = (const float*)d_in[17]; const float* kb2 = (const float*)d_in[18];
    const float* kW3 = (const float*)d_in[19]; const float* kb3 = (const float*)d_in[20];
    const float* eW1 = (const float*)d_in[21]; const float* eb1 = (const float*)d_in[22];
    const float* eW2 = (const float*)d_in[23]; const float* eb2 = (const float*)d_in[24];
    const float* eW3 = (const float*)d_in[25]; const float* eb3 = (const float*)d_in[26];
    const int*   src = (const int*)d_in[27];
    const int*   dst = (const int*)d_in[28];

    const int   N  = in_sizes[1];                 // 2048 waveguides = matrix dim
    const int   E  = in_sizes[27];                // edges
    const int   n  = N;
    const int   e0_off = (in_sizes[0] - N * RESN) / 2;   // (KNN+1)*RES
    const float KWAVE  = 2.0f * 3.14159265358979323846f / 1.55f;
    const float WH     = 0.75f;
    const float CEPS_DX = 1.0f * (1.0f / (float)RESN);   // C_EPSILON * PERIOD/RES

    // workspace: 5 matrix slots + small vectors (~84 MB total)
    float* ws = (float*)d_ws;
    const size_t MS = (size_t)n * n;
    float* s0 = ws;             // C, later T
    float* s1 = ws + 1 * MS;    // Km, later T2
    float* s2 = ws + 2 * MS;    // X (Newton-Schulz / Cinv), later cos
    float* s3 = ws + 3 * MS;    // temps, later sin
    float* s4 = ws + 4 * MS;    // temps
    float* neffs = ws + 5 * MS;
    float* Eys   = neffs + N;
    float* U0    = Eys + (size_t)N * RESN;
    float* UzRe  = U0 + N;
    float* UzIm  = UzRe + N;

    const int eb  = (int)((MS + 255) / 256);      // elementwise grid
    const int nb  = (N + 255) / 256;

    // --- 1. per-waveguide nets ---
    neff_kernel<<<nb, 256, 0, stream>>>(hs, nW1, nb1, nW2, nb2, nW3, nb3, neffs, N);
    eys_kernel<<<nb, 256, 0, stream>>>(hs, eW1, eb1, eW2, eb2, eW3, eb3,
                                       E0, e0_off, Eys, U0, N, CEPS_DX);

    // --- 2. build C = I + scatter(c_e), Km = scatter(k_e) ---
    init_ck_kernel<<<eb, 256, 0, stream>>>(s0, s1, n);
    edge_kernel<<<(E + 255) / 256, 256, 0, stream>>>(hs, dis, src, dst,
        cW1, cb1, cW2, cb2, cW3, cb3, kW1, kb1, kW2, kb2, kW3, kb3,
        s0, s1, N, E, KWAVE);

    // --- 3. Cinv via Newton-Schulz (||I-C|| <= 0.4): X1 = 2I - C, 4 refinements ---
    float* Xp = s2; float* Xt = s4;
    elem_kernel<<<eb, 256, 0, stream>>>(Xp, s0, nullptr, -1.f, 0.f, 2.f, n);
    for (int it = 0; it < 4; ++it) {
        gemm(s0, Xp, nullptr, s3, n, 1.f, 0.f, 0.f, stream);   // Y = C@X
        gemm(Xp, s3, Xp, Xt, n, -1.f, 2.f, 0.f, stream);       // Xn = 2X - X@Y
        float* t = Xp; Xp = Xt; Xt = t;                        // even count -> Xp==s2
    }
    float* Cinv = Xp;   // == s2

    // --- 4. T = (WH/2^8) * Cinv @ (diag(neff*k)*C + Km);  T2 = T@T ---
    m1_kernel<<<eb, 256, 0, stream>>>(s3, s0, s1, neffs, KWAVE, n);     // M1 -> s3
    gemm(Cinv, s3, nullptr, s0, n, WH / 256.0f, 0.f, 0.f, stream);      // T  -> s0
    gemm(s0, s0, nullptr, s1, n, 1.f, 0.f, 0.f, stream);                // T2 -> s1

    // --- 5. cosm/sinm of T (Taylor in T2; ||T|| ~ 0.024) ---
    // cos = ((-T2/720 + I/24)@T2 - I/2)@T2 + I
    elem_kernel<<<eb, 256, 0, stream>>>(s4, s1, nullptr, -1.f / 720.f, 0.f, 1.f / 24.f, n);
    gemm(s4, s1, nullptr, s3, n, 1.f, 0.f, -0.5f, stream);
    gemm(s3, s1, nullptr, s2, n, 1.f, 0.f, 1.0f, stream);               // cos -> s2
    // sin = T @ (((-T2/5040 + I/120)@T2 - I/6)@T2 + I)
    elem_kernel<<<eb, 256, 0, stream>>>(s4, s1, nullptr, -1.f / 5040.f, 0.f, 1.f / 120.f, n);
    gemm(s4, s1, nullptr, s3, n, 1.f, 0.f, -1.f / 6.f, stream);
    gemm(s3, s1, nullptr, s4, n, 1.f, 0.f, 1.0f, stream);
    gemm(s0, s4, nullptr, s3, n, 1.f, 0.f, 0.f, stream);                // sin -> s3

    // --- 6. 8 complex squarings: (X+iY)^2 = (XX - YY) + i(XY + YX) ---
    float *Xc = s2, *Yc = s3, *Xn = s0, *Yn = s1;
    for (int it = 0; it < 8; ++it) {
        gemm(Xc, Xc, nullptr, Xn, n, 1.f, 0.f, 0.f, stream);
        gemm(Yc, Yc, Xn, Xn, n, -1.f, 1.f, 0.f, stream);
        gemm(Xc, Yc, nullptr, Yn, n, 1.f, 0.f, 0.f, stream);
        gemm(Yc, Xc, Yn, Yn, n, 1.f, 1.f, 0.f, stream);
        float* t;
        t = Xc; Xc = Xn; Xn = t;
        t = Yc; Yc = Yn; Yn = t;
    }
    // even iteration count -> P = (Xc, Yc) back in (s2, s3)

    // --- 7. Uz = P @ U0 (U0 real); En = Uz x Eys -> interleaved complex64 ---
    matvec2_kernel<<<(n + 7) / 8, 256, 0, stream>>>(Xc, Yc, U0, UzRe, UzIm, n);
    en_kernel<<<(N * RESN + 255) / 256, 256, 0, stream>>>(UzRe, UzIm, Eys,
                                                          (float*)d_out, N * RESN);

    (void)n_in; (void)out_size; (void)ws_size;
    (void)eb3;  // eb3 (zeros) folded into sb3 staging above
}